// ASTPatternModel_90580860273224
// MI455X (gfx1250) — compile-verified
//
#include <hip/hip_runtime.h>
#include <hip/hip_bf16.h>
#include <math.h>

typedef float v2f __attribute__((ext_vector_type(2)));
typedef float v8f __attribute__((ext_vector_type(8)));

#define D 64

// ---------------------------------------------------------------------------
// 1) embeds = node_emb[nodes] + features @ feat_W.T + feat_b
//    one thread per (node, d) output element
// ---------------------------------------------------------------------------
__global__ void k_init_embeds(const int* __restrict__ nodes,
                              const float* __restrict__ features,   // [N,10]
                              const float* __restrict__ node_emb,   // [VOCAB,64]
                              const float* __restrict__ feat_W,     // [64,10]
                              const float* __restrict__ feat_b,     // [64]
                              float* __restrict__ E, int nNodes) {
  int tid = blockIdx.x * blockDim.x + threadIdx.x;
  if (tid >= nNodes * D) return;
  int i = tid >> 6;
  int d = tid & (D - 1);
  int c = nodes[i];
  float s = feat_b[d] + node_emb[c * D + d];
  const float* __restrict__ frow = features + (size_t)i * 10;
  const float* __restrict__ wrow = feat_W + d * 10;
#pragma unroll
  for (int k = 0; k < 10; ++k) s = fmaf(frow[k], wrow[k], s);
  E[tid] = s;
}

// ---------------------------------------------------------------------------
// 2a) zero the aggregation buffer (float4 stores)
// ---------------------------------------------------------------------------
__global__ void k_zero_f4(float4* __restrict__ p, int n4) {
  int t = blockIdx.x * blockDim.x + threadIdx.x;
  if (t < n4) p[t] = make_float4(0.f, 0.f, 0.f, 0.f);
}

// ---------------------------------------------------------------------------
// 2b) agg[dst] += E[src]   (scatter-add, 16 threads/edge, 4 floats/thread)
//     native global_atomic_add_f32, device scope (RMW resolved at L2)
// ---------------------------------------------------------------------------
__global__ void k_scatter_add(const float* __restrict__ E,
                              const int* __restrict__ edges,  // [nE,2]
                              float* __restrict__ AGG, int nEdges) {
  int tid = blockIdx.x * blockDim.x + threadIdx.x;
  int e = tid >> 4;
  if (e >= nEdges) return;
  int j = (tid & 15) * 4;
  int src = edges[2 * e];
  int dst = edges[2 * e + 1];
  const float4 v = *(const float4*)(E + (size_t)src * D + j);
  float* p = AGG + (size_t)dst * D + j;
  asm volatile(
      "global_atomic_add_f32 %0, %1, off scope:SCOPE_DEV\n\t"
      "global_atomic_add_f32 %0, %2, off offset:4 scope:SCOPE_DEV\n\t"
      "global_atomic_add_f32 %0, %3, off offset:8 scope:SCOPE_DEV\n\t"
      "global_atomic_add_f32 %0, %4, off offset:12 scope:SCOPE_DEV"
      :
      : "v"(p), "v"(v.x), "v"(v.y), "v"(v.z), "v"(v.w)
      : "memory");
}

// ---------------------------------------------------------------------------
// 2c) Y = relu((X + AGG) @ W.T + b)  via V_WMMA_F32_16X16X4_F32
//     8 waves/block; wave = one 16x16 output tile; 2 M-tiles x 4 N-tiles/block
//     A[m][k] = X[m][k]+AGG[m][k];  B[k][n] = W[n][k]  (W row-major [64,64])
// ---------------------------------------------------------------------------
__global__ void k_gemm_bias_relu(const float* __restrict__ X,
                                 const float* __restrict__ AGG,
                                 const float* __restrict__ W,     // conv1_W [64,64]
                                 const float* __restrict__ bias,  // [64]
                                 float* __restrict__ Y) {
  const int lane = threadIdx.x & 31;
  const int wave = threadIdx.x >> 5;               // 0..7
  const int mtile = blockIdx.x * 2 + (wave >> 2);
  const int ntile = wave & 3;
  const int m0 = mtile * 16;
  const int n0 = ntile * 16;
  const int half = lane >> 4;                      // 0: K lo pair, 1: K hi pair
  const int l16 = lane & 15;

  const size_t arow = (size_t)(m0 + l16) * D;      // A row for this lane (M = l16)
  const size_t brow = (size_t)(n0 + l16) * D;      // W row n (B column n = l16)

  v8f acc = {};
#pragma unroll
  for (int kb = 0; kb < 16; ++kb) {
    const int k = kb * 4 + half * 2;
    v2f a = *(const v2f*)(X + arow + k);
    v2f g = *(const v2f*)(AGG + arow + k);
    a.x += g.x;
    a.y += g.y;
    v2f b = *(const v2f*)(W + brow + k);
    acc = __builtin_amdgcn_wmma_f32_16x16x4_f32(
        /*neg_a=*/false, a, /*neg_b=*/false, b,
        /*c_mod=*/(short)0, acc, /*reuse_a=*/false, /*reuse_b=*/false);
  }

  // D layout: VGPR v -> M = v + 8*half (lanes 16-31 hold M=8..15), N = l16
  const int n = n0 + l16;
  const float bn = bias[n];
#pragma unroll
  for (int v = 0; v < 8; ++v) {
    const int m = m0 + v + half * 8;
    float r = acc[v] + bn;
    Y[(size_t)m * D + n] = r > 0.f ? r : 0.f;
  }
}

// ---------------------------------------------------------------------------
// 3) cosine similarity vs pattern_emb[pattern_id]; one wave32 per node
// ---------------------------------------------------------------------------
__global__ void k_cosine(const float* __restrict__ E,
                         const float* __restrict__ pattern_emb,
                         const int* __restrict__ pattern_id,
                         float* __restrict__ out, int nNodes) {
  int gtid = blockIdx.x * blockDim.x + threadIdx.x;
  int node = gtid >> 5;
  int lane = threadIdx.x & 31;
  if (node >= nNodes) return;
  const float* __restrict__ p = pattern_emb + (size_t)pattern_id[0] * D;
  float x0 = E[(size_t)node * D + lane];
  float x1 = E[(size_t)node * D + lane + 32];
  float p0 = p[lane];
  float p1 = p[lane + 32];
  float num = x0 * p0 + x1 * p1;
  float ss = x0 * x0 + x1 * x1;
  float pp = p0 * p0 + p1 * p1;
#pragma unroll
  for (int off = 16; off; off >>= 1) {
    num += __shfl_xor(num, off, 32);
    ss += __shfl_xor(ss, off, 32);
    pp += __shfl_xor(pp, off, 32);
  }
  if (lane == 0) {
    float denom = fmaxf(sqrtf(ss), 1e-8f) * fmaxf(sqrtf(pp), 1e-8f);
    out[node] = num / denom;
  }
}

// ---------------------------------------------------------------------------
extern "C" void kernel_launch(void* const* d_in, const int* in_sizes, int n_in,
                              void* d_out, int out_size, void* d_ws, size_t ws_size,
                              hipStream_t stream) {
  const int* nodes        = (const int*)d_in[0];
  const int* edges        = (const int*)d_in[1];
  const float* features   = (const float*)d_in[2];
  const float* node_emb   = (const float*)d_in[3];
  const float* feat_W     = (const float*)d_in[4];
  const float* feat_b     = (const float*)d_in[5];
  const float* conv1_W    = (const float*)d_in[6];
  const float* conv1_b    = (const float*)d_in[7];
  const float* pattern_emb= (const float*)d_in[8];
  const int* pattern_id   = (const int*)d_in[9];
  float* out = (float*)d_out;

  const int nNodes = in_sizes[0];          // 100000
  const int nEdges = in_sizes[1] / 2;      // 800000

  float* E0  = (float*)d_ws;
  float* E1  = E0 + (size_t)nNodes * D;
  float* AGG = E1 + (size_t)nNodes * D;

  const int threads = 256;
  const int gInit    = (nNodes * D + threads - 1) / threads;
  const int n4       = nNodes * D / 4;
  const int gZero    = (n4 + threads - 1) / threads;
  const int gScatter = (nEdges * 16 + threads - 1) / threads;
  const int gGemm    = nNodes / 32;        // 2 M-tiles of 16 rows per block
  const int gCos     = (nNodes * 32 + threads - 1) / threads;

  // embeds = node_emb[nodes] + features @ feat_W.T + feat_b
  k_init_embeds<<<gInit, threads, 0, stream>>>(nodes, features, node_emb,
                                               feat_W, feat_b, E0, nNodes);
  // round 1
  k_zero_f4<<<gZero, threads, 0, stream>>>((float4*)AGG, n4);
  k_scatter_add<<<gScatter, threads, 0, stream>>>(E0, edges, AGG, nEdges);
  k_gemm_bias_relu<<<gGemm, threads, 0, stream>>>(E0, AGG, conv1_W, conv1_b, E1);
  // round 2
  k_zero_f4<<<gZero, threads, 0, stream>>>((float4*)AGG, n4);
  k_scatter_add<<<gScatter, threads, 0, stream>>>(E1, edges, AGG, nEdges);
  k_gemm_bias_relu<<<gGemm, threads, 0, stream>>>(E1, AGG, conv1_W, conv1_b, E0);
  // cosine vs pattern
  k_cosine<<<gCos, threads, 0, stream>>>(E0, pattern_emb, pattern_id, out, nNodes);
}